// HalfWarper_87076166959371
// MI455X (gfx1250) — compile-verified
//
#include <hip/hip_runtime.h>
#include <math.h>

// ---------------- problem constants (from setup_inputs) ----------------
constexpr int Bb = 4;
constexpr int H  = 544;
constexpr int W  = 960;
constexpr int HW = H * W;          // 522240
constexpr int Np = Bb * HW;        // 2088960 pixels per image set

typedef __attribute__((ext_vector_type(2))) float v2f;
typedef __attribute__((ext_vector_type(8))) float v8f;

// ---------------- helpers ----------------
__device__ __forceinline__ float srgb_lin(float v) {
    return v > 0.04045f ? powf((v + 0.055f) * (1.0f / 1.055f), 2.4f)
                        : v * (1.0f / 12.92f);
}
__device__ __forceinline__ float lab_f(float t) {
    return t > 0.008856f ? cbrtf(fmaxf(t, 1e-8f))
                         : 7.787f * t + (4.0f / 29.0f);
}

// ---------------- Kernel 1: RGB -> Lab, 3x3 einsum via V_WMMA_F32_16X16X4_F32
// One wave processes 32 consecutive pixels of [I0 | I1] with two WMMAs.
// B-matrix layout (K=0 row -> v0 lanes0-15, K=2 -> v0 lanes16-31,
// K=1 -> v1 lanes0-15, K=3 -> v1 lanes16-31) means every lane loads exactly
// the planar R/G/B values it owns: no shuffles on the input side.
__global__ void lab_kernel(const float* __restrict__ I0,
                           const float* __restrict__ I1,
                           float* __restrict__ lab /* lab0:3N, lab1:3N */) {
    const int t    = blockIdx.x * blockDim.x + threadIdx.x;   // 0 .. 2N-1
    const int lane = threadIdx.x & 31;
    const int wbase = t - lane;                               // wave's first pixel id

    const float* img;
    float* outp;
    int q0;
    if (wbase < Np) { img = I0; outp = lab;             q0 = wbase;      }
    else            { img = I1; outp = lab + 3 * (size_t)Np; q0 = wbase - Np; }

    const int  row = lane & 15;
    const bool hi  = lane >= 16;

    // A(16x4) = color matrix M, rows 3..15 and col 3 zero (branch-free selects)
    float m0r = row == 0 ? 0.412453f : (row == 1 ? 0.212671f : (row == 2 ? 0.019334f : 0.0f));
    float m1r = row == 0 ? 0.357580f : (row == 1 ? 0.715160f : (row == 2 ? 0.119193f : 0.0f));
    float m2r = row == 0 ? 0.180423f : (row == 1 ? 0.072169f : (row == 2 ? 0.950227f : 0.0f));
    v2f A;
    A.x = hi ? m2r : m0r;     // v0: K=0 (low half) / K=2 (high half)
    A.y = hi ? 0.f : m1r;     // v1: K=1 (low half) / K=3 = 0 (high half)

    // column pixels this lane supplies
    const int qa = q0 + row;        // WMMA #1 columns: pixels q0..q0+15
    const int qb = q0 + 16 + row;   // WMMA #2 columns: pixels q0+16..q0+31

    auto plane = [&](int q, int c) -> float {
        int b = q / HW, s = q - b * HW;
        return img[(size_t)b * 3 * HW + (size_t)c * HW + s];
    };

    const int cA = hi ? 2 : 0;      // low half supplies R (K=0), high half supplies B (K=2)
    float v0a = plane(qa, cA), v1a = plane(qa, 1);
    float v0b = plane(qb, cA), v1b = plane(qb, 1);

    v2f B1, B2;
    B1.x = srgb_lin(v0a);  B1.y = hi ? 0.f : srgb_lin(v1a);   // K=1 row (G); K=3 row = 0
    B2.x = srgb_lin(v0b);  B2.y = hi ? 0.f : srgb_lin(v1b);

    v8f C = {};
    v8f D1 = __builtin_amdgcn_wmma_f32_16x16x4_f32(false, A, false, B1, (short)0, C, false, false);
    v8f D2 = __builtin_amdgcn_wmma_f32_16x16x4_f32(false, A, false, B2, (short)0, C, false, false);

    // D rows 0..2 live in lanes 0-15 of d[0..2]; hand WMMA#2 rows to lanes 16-31
    float sx = __shfl(D2[0], row, 32);
    float sy = __shfl(D2[1], row, 32);
    float sz = __shfl(D2[2], row, 32);
    float X = hi ? sx : D1[0];
    float Y = hi ? sy : D1[1];
    float Z = hi ? sz : D1[2];

    float fx = lab_f(X * (1.0f / 0.95047f));
    float fy = lab_f(Y);
    float fz = lab_f(Z * (1.0f / 1.08883f));

    const int q = q0 + lane;
    const int b = q / HW, s = q - b * HW;
    float* o = outp + (size_t)b * 3 * HW + s;
    o[0]        = 116.0f * fy - 16.0f;
    o[HW]       = 500.0f * (fx - fy);
    o[2 * HW]   = 200.0f * (fy - fz);
}

// ---------------- Kernel 2: metric (backward-warp gather) + softsplat scatter
__global__ void splat_kernel(const float* __restrict__ img,
                             const float* __restrict__ flow,
                             const float* __restrict__ labSelf,
                             const float* __restrict__ labOther,
                             float* __restrict__ acc /* (B,4,HW) */) {
    const int t = blockIdx.x * blockDim.x + threadIdx.x;      // 0 .. N-1
    const int b = t / HW, s = t - b * HW;
    const int i = s / W,  j = s - i * W;

    const float fy = flow[(size_t)b * 2 * HW + s];
    const float fx = flow[(size_t)b * 2 * HW + HW + s];

    // ---- backward warp sample of labOther (border-clamped bilinear)
    float gy = 2.0f * fy / H + (-1.0f + 2.0f * i / (H - 1));
    float gx = 2.0f * fx / W + (-1.0f + 2.0f * j / (W - 1));
    float xs = ((gx + 1.0f) * W - 1.0f) * 0.5f;
    float ys = ((gy + 1.0f) * H - 1.0f) * 0.5f;
    float x0 = floorf(xs), y0 = floorf(ys);
    float wx = xs - x0,    wy = ys - y0;
    int x0i = (int)fminf(fmaxf(x0,        0.f), (float)(W - 1));
    int x1i = (int)fminf(fmaxf(x0 + 1.0f, 0.f), (float)(W - 1));
    int y0i = (int)fminf(fmaxf(y0,        0.f), (float)(H - 1));
    int y1i = (int)fminf(fmaxf(y0 + 1.0f, 0.f), (float)(H - 1));

    const float* Lo = labOther + (size_t)b * 3 * HW;
    const float* Ls = labSelf  + (size_t)b * 3 * HW;
    float dd = 0.0f;
    #pragma unroll
    for (int c = 0; c < 3; ++c) {
        const float* pl = Lo + (size_t)c * HW;
        float v00 = pl[y0i * W + x0i], v01 = pl[y0i * W + x1i];
        float v10 = pl[y1i * W + x0i], v11 = pl[y1i * W + x1i];
        float samp = v00 * (1.f - wx) * (1.f - wy) + v01 * wx * (1.f - wy)
                   + v10 * (1.f - wx) * wy         + v11 * wx * wy;
        float d = Ls[(size_t)c * HW + s] - samp;
        dd += d * d;
    }
    const float m = expf(-0.1f * sqrtf(dd));

    // ---- forward splat of [rgb*m, m] at (j+fx, i+fy)
    const float r  = img[(size_t)b * 3 * HW + s];
    const float g  = img[(size_t)b * 3 * HW + HW + s];
    const float bl = img[(size_t)b * 3 * HW + 2 * HW + s];

    float Xc = (float)j + fx, Yc = (float)i + fy;
    float xf0 = floorf(Xc), yf0 = floorf(Yc);
    int xi0 = (int)xf0, yi0 = (int)yf0;
    float* A = acc + (size_t)b * 4 * HW;

    #pragma unroll
    for (int dy = 0; dy <= 1; ++dy) {
        #pragma unroll
        for (int dx = 0; dx <= 1; ++dx) {
            int xi = xi0 + dx, yi = yi0 + dy;
            float w = (1.0f - fabsf(Xc - (xf0 + dx))) * (1.0f - fabsf(Yc - (yf0 + dy)));
            if (xi >= 0 && xi < W && yi >= 0 && yi < H && w != 0.0f) {
                int idx = yi * W + xi;
                float wm = w * m;
                atomicAdd(&A[idx],            r  * wm);
                atomicAdd(&A[HW + idx],       g  * wm);
                atomicAdd(&A[2 * HW + idx],   bl * wm);
                atomicAdd(&A[3 * HW + idx],   wm);
            }
        }
    }
}

// ---------------- Kernels 3a-3d: separable 5x5 morphological open on indicator
// erode = min-pool (+inf pad -> skip OOB), dilate = max-pool (-inf pad -> skip OOB)
__global__ void rowmin_ind_kernel(const float* __restrict__ acc0,
                                  const float* __restrict__ acc1,
                                  const int* __restrict__ kptr,
                                  float* __restrict__ out /* 2N */) {
    const int t = blockIdx.x * blockDim.x + threadIdx.x;      // 0 .. 2N-1
    const int dir = t / Np, p = t - dir * Np;
    const int b = p / HW, s = p - b * HW;
    const int i = s / W,  j = s - i * W;
    const int rr = kptr[0] >> 1;
    const float* sp = (dir ? acc1 : acc0) + (size_t)b * 4 * HW + 3 * (size_t)HW + (size_t)i * W;
    float mv = INFINITY;
    for (int dj = -rr; dj <= rr; ++dj) {
        int jj = j + dj;
        if (jj >= 0 && jj < W) {
            float ind = (sp[jj] != 0.0f) ? 1.0f : 0.0f;
            mv = fminf(mv, ind);
        }
    }
    out[t] = mv;
}

__global__ void colpool_kernel(const float* __restrict__ in,
                               const int* __restrict__ kptr,
                               float* __restrict__ out, int isMax) {
    const int t = blockIdx.x * blockDim.x + threadIdx.x;      // 0 .. 2N-1
    const int dir = t / Np, p = t - dir * Np;
    const int b = p / HW, s = p - b * HW;
    const int i = s / W,  j = s - i * W;
    const int rr = kptr[0] >> 1;
    const float* col = in + (size_t)dir * Np + (size_t)b * HW;
    float mv = isMax ? -INFINITY : INFINITY;
    for (int di = -rr; di <= rr; ++di) {
        int ii = i + di;
        if (ii >= 0 && ii < H) {
            float v = col[ii * W + j];
            mv = isMax ? fmaxf(mv, v) : fminf(mv, v);
        }
    }
    out[t] = mv;
}

__global__ void rowmax_kernel(const float* __restrict__ in,
                              const int* __restrict__ kptr,
                              float* __restrict__ out) {
    const int t = blockIdx.x * blockDim.x + threadIdx.x;      // 0 .. 2N-1
    const int dir = t / Np, p = t - dir * Np;
    const int b = p / HW, s = p - b * HW;
    const int i = s / W,  j = s - i * W;
    const int rr = kptr[0] >> 1;
    const float* rowp = in + (size_t)dir * Np + (size_t)b * HW + (size_t)i * W;
    float mv = -INFINITY;
    for (int dj = -rr; dj <= rr; ++dj) {
        int jj = j + dj;
        if (jj >= 0 && jj < W) mv = fmaxf(mv, rowp[jj]);
    }
    out[t] = mv;
}

// ---------------- Kernel 4: normalize accumulators + blend + write outputs
__global__ void blend_kernel(const float* __restrict__ acc0,
                             const float* __restrict__ acc1,
                             const float* __restrict__ mo /* m0:N, m1:N */,
                             float* __restrict__ out /* base0:4N, base1:4N */) {
    const int t = blockIdx.x * blockDim.x + threadIdx.x;      // 0 .. N-1
    const int b = t / HW, s = t - b * HW;

    const float* A0 = acc0 + (size_t)b * 4 * HW;
    const float* A1 = acc1 + (size_t)b * 4 * HW;
    float s0 = A0[3 * HW + s], s1 = A1[3 * HW + s];
    float n0 = (s0 == 0.0f) ? 1.0f : s0;
    float n1 = (s1 == 0.0f) ? 1.0f : s1;
    float m0 = mo[(size_t)b * HW + s];
    float m1 = mo[(size_t)Np + (size_t)b * HW + s];

    float* O0 = out + (size_t)b * 4 * HW;
    float* O1 = out + (size_t)4 * Np + (size_t)b * 4 * HW;
    #pragma unroll
    for (int c = 0; c < 3; ++c) {
        float w0 = A0[(size_t)c * HW + s] / n0;
        float w1 = A1[(size_t)c * HW + s] / n1;
        O0[(size_t)c * HW + s] = m0 * w0 + (1.0f - m0) * w1;
        O1[(size_t)c * HW + s] = m1 * w1 + (1.0f - m1) * w0;
    }
    O0[3 * (size_t)HW + s] = m0;
    O1[3 * (size_t)HW + s] = m1;
}

// ---------------- launcher ----------------
extern "C" void kernel_launch(void* const* d_in, const int* in_sizes, int n_in,
                              void* d_out, int out_size, void* d_ws, size_t ws_size,
                              hipStream_t stream) {
    const float* I0   = (const float*)d_in[0];
    const float* I1   = (const float*)d_in[1];
    const float* fl1  = (const float*)d_in[2];   // flow1tot
    const float* fl0  = (const float*)d_in[3];   // flow0tot
    const int*   kptr = (const int*)d_in[4];

    float* lab  = (float*)d_ws;                  // 6N floats: lab0 | lab1
    float* acc0 = lab + (size_t)6 * Np;          // 4N
    float* acc1 = acc0 + (size_t)4 * Np;         // 4N
    // pooling buffers reuse the lab region after the splat pass
    float* bufA = lab;                           // 2N
    float* bufB = lab + (size_t)2 * Np;          // 2N
    float* bufC = lab + (size_t)4 * Np;          // 2N  (m0 | m1)
    float* out  = (float*)d_out;

    const int TB = 256;

    // 1) RGB -> Lab for both images (WMMA einsum)
    lab_kernel<<<(2 * Np) / TB, TB, 0, stream>>>(I0, I1, lab);

    // 2) zero splat accumulators (capture-safe memset node)
    hipMemsetAsync(acc0, 0, (size_t)8 * Np * sizeof(float), stream);

    // 3) metric + softsplat, both directions
    //    dir 0: splat I0 with flow0tot, metric from lab0 - warp(lab1, flow0tot)
    splat_kernel<<<Np / TB, TB, 0, stream>>>(I0, fl0, lab,                 lab + 3 * (size_t)Np, acc0);
    //    dir 1: splat I1 with flow1tot, metric from lab1 - warp(lab0, flow1tot)
    splat_kernel<<<Np / TB, TB, 0, stream>>>(I1, fl1, lab + 3 * (size_t)Np, lab,                 acc1);

    // 4) morphological open (erode then dilate), separable passes
    rowmin_ind_kernel<<<(2 * Np) / TB, TB, 0, stream>>>(acc0, acc1, kptr, bufA);
    colpool_kernel  <<<(2 * Np) / TB, TB, 0, stream>>>(bufA, kptr, bufB, /*isMax=*/0);
    rowmax_kernel   <<<(2 * Np) / TB, TB, 0, stream>>>(bufB, kptr, bufA);
    colpool_kernel  <<<(2 * Np) / TB, TB, 0, stream>>>(bufA, kptr, bufC, /*isMax=*/1);

    // 5) normalize + blend + emit (base0, base1)
    blend_kernel<<<Np / TB, TB, 0, stream>>>(acc0, acc1, bufC, out);
}